// PGA_Unfold_J10_PC_10900626997257
// MI455X (gfx1250) — compile-verified
//
#include <hip/hip_runtime.h>

namespace {

constexpr int KB  = 4;    // K
constexpr int TT  = 128;  // T
constexpr int NT  = 64;
constexpr int NRF = 8;
constexpr int MM  = 8;
constexpr float SIGMA2 = 1.0f;
constexpr float LOG2F_ = 0.69314718055994530942f;
constexpr float PT     = 8.0f;

typedef __attribute__((ext_vector_type(2))) float v2f;
typedef __attribute__((ext_vector_type(8))) float v8f;

__device__ __forceinline__ v8f wmma4(v2f a, v2f b, v8f c) {
  // V_WMMA_F32_16X16X4_F32 : D = A(16x4) * B(4x16) + C(16x16)
  return __builtin_amdgcn_wmma_f32_16x16x4_f32(false, a, false, b, (short)0, c,
                                               false, false);
}

__device__ __forceinline__ float2 cmul(float2 a, float2 b) {
  return make_float2(a.x * b.x - a.y * b.y, a.x * b.y + a.y * b.x);
}
__device__ __forceinline__ float2 cadd(float2 a, float2 b) {
  return make_float2(a.x + b.x, a.y + b.y);
}
__device__ __forceinline__ float2 cconj(float2 a) { return make_float2(a.x, -a.y); }
__device__ __forceinline__ float finz(float x) { return __builtin_isfinite(x) ? x : 0.0f; }
__device__ __forceinline__ float2 clampmag(float2 d, float mx) {
  float mag = sqrtf(d.x * d.x + d.y * d.y + 1e-24f);
  float sc  = fminf(mx / (mag + 1e-12f), 1.0f);
  return make_float2(d.x * sc, d.y * sc);
}

struct Smem {
  float2 E[NT][NT];        // 32 KB : 64x64 complex scratch (E = A A^H - R)
  float2 G[NT][16];        //  8 KB : GEMM output; first 64x8 aliased as A
  float2 C[NT][16];        //  8 KB : padded B operand (A W^H, A, ...)
  float2 Hs[NT][MM];       //  4 KB : H[k,t] tile
  float2 Bd[MM][16];       //  1 KB : B1-B2 rows (padded to 16 cols)
  float2 f[KB][NT];        //  2 KB : diagonal of F per k (pc-masked entries)
  float2 Wm[KB][NRF][MM];  //  2 KB
  float2 am[MM][NRF];      // a_m = F^H h_m
  float2 V[NRF][NRF];      // V = W W^H
  float2 gFc[NT];
  float2 gFr[NT];
  float  red[256];
  float  tr1[MM];
  float2 sm[MM];
  float  wr2[NRF];
  float  rn2[KB];
};

__device__ float block_reduce(Smem& s, float v, int tid) {
  __syncthreads();
  s.red[tid] = v;
  __syncthreads();
  for (int st = 128; st > 0; st >>= 1) {
    if (tid < st) s.red[tid] += s.red[tid + st];
    __syncthreads();
  }
  float r = s.red[0];
  __syncthreads();
  return r;
}

// Complex C(MxN) = A(MxK) * op(B), op(B) = B (KxN) or conj(B)^T with B (NxK).
// All row-major float2 in LDS, ld in float2 units; M,N multiples of 16, K of 4.
// Each wave owns whole 16x16 tiles -> EXEC is all-ones for every WMMA.
__device__ void cgemm(const float2* A, int lda, const float2* B, int ldb,
                      bool bConjT, float2* C, int ldc, int Mdim, int Ndim,
                      int Kdim, int tid) {
  const int wave = tid >> 5, lane = tid & 31;
  const int l15 = lane & 15, h = lane >> 4;
  const int tn = Ndim >> 4;
  const int ntiles = (Mdim >> 4) * tn;
  for (int tile = wave; tile < ntiles; tile += 8) {
    const int ti = tile / tn, tj = tile - ti * tn;
    const int m = ti * 16 + l15, n = tj * 16 + l15;
    v8f cr = {};
    v8f ci = {};
    for (int k0 = 0; k0 < Kdim; k0 += 4) {
      const int ka = k0 + 2 * h;  // A frag: lanes0-15 K={k0,k0+1}, lanes16-31 K={k0+2,k0+3}
      float2 a0 = A[m * lda + ka];
      float2 a1 = A[m * lda + ka + 1];
      float2 b0, b1;
      if (bConjT) {
        b0 = B[n * ldb + ka];
        b1 = B[n * ldb + ka + 1];
        b0.y = -b0.y;
        b1.y = -b1.y;
      } else {
        b0 = B[ka * ldb + n];
        b1 = B[(ka + 1) * ldb + n];
      }
      v2f ar  = {a0.x, a1.x};
      v2f ai  = {a0.y, a1.y};
      v2f nai = {-a0.y, -a1.y};
      v2f br  = {b0.x, b1.x};
      v2f bi  = {b0.y, b1.y};
      cr = wmma4(ar, br, cr);    // Re += ArBr
      cr = wmma4(nai, bi, cr);   // Re -= AiBi
      ci = wmma4(ar, bi, ci);    // Im += ArBi
      ci = wmma4(ai, br, ci);    // Im += AiBr
    }
#pragma unroll
    for (int v = 0; v < 8; v++) {
      const int mm = ti * 16 + v + 8 * h;  // C/D layout: vgpr v, halves offset by 8
      C[mm * ldc + n] = make_float2(cr[v], ci[v]);
    }
  }
}

__device__ void load_H(Smem& s, const float2* Hg, int k, int t, int tid) {
  const float2* src = Hg + (size_t)(k * TT + t) * (NT * MM);
  for (int i = tid; i < NT * MM; i += 256) (&s.Hs[0][0])[i] = src[i];
}

// V = W W^H, a_m = F^H h_m, tr1_m = a_m^H V a_m, s_m = a_m^H w_m
__device__ void compute_stats(Smem& s, int k, int tid) {
  __syncthreads();
  if (tid < 64) {
    int r = tid >> 3, c = tid & 7;
    float2 acc = make_float2(0.f, 0.f);
    for (int m = 0; m < MM; m++)
      acc = cadd(acc, cmul(s.Wm[k][r][m], cconj(s.Wm[k][c][m])));
    s.V[r][c] = acc;
  } else if (tid < 128) {
    int q = tid - 64;
    int m = q >> 3, r = q & 7;
    float2 acc = make_float2(0.f, 0.f);
    for (int i = 0; i < 8; i++)
      acc = cadd(acc, cmul(cconj(s.f[k][8 * r + i]), s.Hs[8 * r + i][m]));
    s.am[m][r] = acc;
  }
  __syncthreads();
  if (tid < 8) {
    int m = tid;
    float tr = 0.f;
    float2 sm = make_float2(0.f, 0.f);
    for (int r = 0; r < 8; r++) {
      float2 t2 = make_float2(0.f, 0.f);
      for (int c = 0; c < 8; c++) t2 = cadd(t2, cmul(s.V[r][c], s.am[m][c]));
      float2 pr = cmul(cconj(s.am[m][r]), t2);
      tr += pr.x;
      sm = cadd(sm, cmul(cconj(s.am[m][r]), s.Wm[k][r][m]));
    }
    s.tr1[m] = tr;
    s.sm[m]  = sm;
  }
  __syncthreads();
}

// A = F@W, exploiting pc sparsity: A[n,:] = f_n * W[n/8,:]. Stored in G[0..511].
__device__ void build_A(Smem& s, int k, int tid) {
  float2* A = &s.G[0][0];
  for (int i = tid; i < NT * MM; i += 256) {
    int n = i >> 3, m = i & 7;
    A[i] = cmul(s.f[k][n], s.Wm[k][n >> 3][m]);
  }
}

// ||F@W||_F = sqrt( sum_n |f_n|^2 * ||W[n/8,:]||^2 )
__device__ float compute_p(Smem& s, int k, int tid) {
  __syncthreads();
  if (tid < 8) {
    float acc = 0.f;
    for (int m = 0; m < MM; m++) {
      float2 w = s.Wm[k][tid][m];
      acc += w.x * w.x + w.y * w.y;
    }
    s.wr2[tid] = acc;
  }
  __syncthreads();
  float v = 0.f;
  if (tid < 64) {
    float2 fv = s.f[k][tid];
    v = (fv.x * fv.x + fv.y * fv.y) * s.wr2[tid >> 3];
  }
  return sqrtf(block_reduce(s, v, tid));
}

__device__ void eval_rate_tau(Smem& s, const float2* Hg, const float2* Rg, int t,
                              int tid, float* rate_out, float* tau_out) {
  float rate = 0.f, tau = 0.f;
  for (int k = 0; k < KB; k++) {
    load_H(s, Hg, k, t, tid);
    compute_stats(s, k, tid);
    float v = 0.f;
    if (tid < 8) {
      float t1 = s.tr1[tid];
      float2 sm = s.sm[tid];
      float t2v = t1 - (sm.x * sm.x + sm.y * sm.y);
      v = (logf(t1 + SIGMA2) - logf(t2v + SIGMA2)) / LOG2F_;
    }
    rate += block_reduce(s, v, tid);
    build_A(s, k, tid);
    __syncthreads();
    cgemm(&s.G[0][0], 8, &s.G[0][0], 8, true, &s.E[0][0], NT, NT, NT, 8, tid);
    __syncthreads();
    const float2* Rk = Rg + (size_t)k * NT * NT;
    float p = 0.f;
    for (int i = tid; i < NT * NT; i += 256) {
      float2 e = (&s.E[0][0])[i];
      float2 r = Rk[i];
      float ex = e.x - r.x, ey = e.y - r.y;
      p += ex * ex + ey * ey;
    }
    tau += block_reduce(s, p, tid) / s.rn2[k];
  }
  *rate_out = rate * 0.25f;
  *tau_out  = tau * 0.25f;
}

__global__ __launch_bounds__(256) void pga_unfold_kernel(
    const float2* __restrict__ Hg, const float2* __restrict__ Rg,
    const float2* __restrict__ F0g, const float2* __restrict__ W0g,
    const float* __restrict__ stepg, float* __restrict__ out) {
  __shared__ Smem s;
  const int t   = blockIdx.x;
  const int tid = threadIdx.x;

  if (tid == 0) __builtin_prefetch(Rg, 0, 3);

  // ---- init: F = sanitize(F0*pc), W = W0 ----
  {
    int k = tid >> 6, idx = tid & 63;
    float2 f0 = F0g[((size_t)(k * TT + t) * NT + idx) * NRF + (idx >> 3)];
    s.f[k][idx] = make_float2(finz(f0.x), finz(f0.y));
    float2 w0 = W0g[(size_t)(k * TT + t) * (NRF * MM) + idx];
    s.Wm[k][idx >> 3][idx & 7] = w0;
  }
  __syncthreads();

  // ---- rn2[k] = ||R_k||_F^2 ----
  for (int k = 0; k < KB; k++) {
    const float2* Rk = Rg + (size_t)k * NT * NT;
    float p = 0.f;
    for (int i = tid; i < NT * NT; i += 256) {
      float2 r = Rk[i];
      p += r.x * r.x + r.y * r.y;
    }
    float tot = block_reduce(s, p, tid);
    if (tid == 0) s.rn2[k] = tot;
    __syncthreads();
  }

  // ---- normalize(F, W0): W *= sqrt(PT)/(||F@W||+eps) ----
  for (int k = 0; k < KB; k++) {
    float p  = compute_p(s, k, tid);
    float sc = sqrtf(PT) / (p + 1e-12f);
    if (tid < 64) {
      float2 w = s.Wm[k][tid >> 3][tid & 7];
      s.Wm[k][tid >> 3][tid & 7] = make_float2(finz(w.x * sc), finz(w.y * sc));
    }
    __syncthreads();
  }

  float rates[3], taus[3];
  eval_rate_tau(s, Hg, Rg, t, tid, &rates[0], &taus[0]);

  for (int ii = 0; ii < 2; ii++) {
    // ================= inner F steps =================
    for (int jj = 0; jj < 2; jj++) {
      if (tid < 64) {
        s.gFc[tid] = make_float2(0, 0);
        s.gFr[tid] = make_float2(0, 0);
      }
      __syncthreads();
      for (int k = 0; k < KB; k++) {
        load_H(s, Hg, k, t, tid);
        compute_stats(s, k, tid);
        // Bd[m,:] = (a_m^H V)/d1 - (a_m^H V - s_m w_m^H)/d2
        if (tid < 64) {
          int m = tid >> 3, j = tid & 7;
          float2 q = make_float2(0, 0);
          for (int r = 0; r < 8; r++)
            q = cadd(q, cmul(cconj(s.am[m][r]), s.V[r][j]));
          float2 smv = s.sm[m];
          float s2 = smv.x * smv.x + smv.y * smv.y;
          float d1 = LOG2F_ * (s.tr1[m] + SIGMA2) + 1e-4f;
          float d2 = LOG2F_ * (s.tr1[m] - s2 + SIGMA2) + 1e-4f;
          float2 t2 = cmul(smv, cconj(s.Wm[k][j][m]));
          s.Bd[m][j] = make_float2(q.x / d1 - (q.x - t2.x) / d2,
                                   q.y / d1 - (q.y - t2.y) / d2);
        } else if (tid < 128) {
          int q2 = tid - 64;
          s.Bd[q2 >> 3][(q2 & 7) + 8] = make_float2(0, 0);  // pad cols 8..15
        }
        __syncthreads();
        // gFc_k = H @ Bd   (sum over m inside the GEMM)
        cgemm(&s.Hs[0][0], 8, &s.Bd[0][0], 16, false, &s.G[0][0], 16, 64, 16, 8, tid);
        __syncthreads();
        if (tid < 64) s.gFc[tid] = cadd(s.gFc[tid], s.G[tid][tid >> 3]);
        __syncthreads();
        // gFr_k = (2/rn2) * (A A^H - R) @ (A W^H)
        build_A(s, k, tid);  // overwrites G[0..511] (dead)
        for (int i = tid; i < 64 * 16; i += 256) {
          int n = i >> 4, m2 = i & 15;
          s.C[n][m2] = (m2 < 8) ? cmul(s.f[k][n], s.V[n >> 3][m2])
                                : make_float2(0, 0);  // A W^H = f_n * V[n/8,:]
        }
        __syncthreads();
        cgemm(&s.G[0][0], 8, &s.G[0][0], 8, true, &s.E[0][0], 64, 64, 64, 8, tid);
        __syncthreads();
        {
          const float2* Rk = Rg + (size_t)k * NT * NT;
          for (int i = tid; i < NT * NT; i += 256) {
            float2 e = (&s.E[0][0])[i];
            float2 r = Rk[i];
            (&s.E[0][0])[i] = make_float2(e.x - r.x, e.y - r.y);
          }
        }
        __syncthreads();
        cgemm(&s.E[0][0], 64, &s.C[0][0], 16, false, &s.G[0][0], 16, 64, 16, 64, tid);
        __syncthreads();
        if (tid < 64) {
          float2 g = s.G[tid][tid >> 3];
          float sc = 2.0f / s.rn2[k];
          s.gFr[tid] = cadd(s.gFr[tid], make_float2(g.x * sc, g.y * sc));
        }
        __syncthreads();
      }
      // F update (same delta for every k), then per-k power renorm
      float ssF = stepg[jj * 10 + ii * 5 + 0];
      if (tid < 64) {
        float2 gc = s.gFc[tid], gr = s.gFr[tid];
        float2 dc = clampmag(make_float2(ssF * gc.x * 0.25f, ssF * gc.y * 0.25f), 0.5f);
        float2 dr = clampmag(make_float2(ssF * gr.x * 0.25f, ssF * gr.y * 0.25f), 0.5f);
        s.gFc[tid] = make_float2(dc.x - dr.x, dc.y - dr.y);
      }
      __syncthreads();
      {
        int k = tid >> 6, n = tid & 63;
        float2 fv = s.f[k][n];
        float2 d  = s.gFc[n];
        s.f[k][n] = make_float2(finz(fv.x + d.x), finz(fv.y + d.y));
      }
      __syncthreads();
      for (int k = 0; k < KB; k++) {
        float p  = compute_p(s, k, tid);
        float sc = sqrtf(PT) / (p + 1e-12f);
        if (tid < 64) {
          float2 fv = s.f[k][tid];
          s.f[k][tid] = make_float2(finz(fv.x * sc), finz(fv.y * sc));
        }
        __syncthreads();
      }
    }
    // ---- project unit modulus (masked entries only; tiny -> 1+0j) ----
    {
      int k = tid >> 6, n = tid & 63;
      float2 fv = s.f[k][n];
      float mag = sqrtf(fv.x * fv.x + fv.y * fv.y);
      float2 o = (mag > 1e-12f) ? make_float2(fv.x / mag, fv.y / mag)
                                : make_float2(1.f, 0.f);
      s.f[k][n] = make_float2(finz(o.x), finz(o.y));
    }
    __syncthreads();
    // ================= W step (per k) =================
    for (int k = 0; k < KB; k++) {
      load_H(s, Hg, k, t, tid);
      compute_stats(s, k, tid);
      build_A(s, k, tid);
      __syncthreads();
      cgemm(&s.G[0][0], 8, &s.G[0][0], 8, true, &s.E[0][0], 64, 64, 64, 8, tid);
      __syncthreads();
      {
        const float2* Rk = Rg + (size_t)k * NT * NT;
        for (int i = tid; i < NT * NT; i += 256) {
          float2 e = (&s.E[0][0])[i];
          float2 r = Rk[i];
          (&s.E[0][0])[i] = make_float2(e.x - r.x, e.y - r.y);
        }
        for (int i = tid; i < 64 * 16; i += 256) {  // C = A padded to 16 cols
          int n = i >> 4, m2 = i & 15;
          s.C[n][m2] = (m2 < 8) ? (&s.G[0][0])[n * 8 + m2] : make_float2(0, 0);
        }
      }
      __syncthreads();
      cgemm(&s.E[0][0], 64, &s.C[0][0], 16, false, &s.G[0][0], 16, 64, 16, 64, tid);
      __syncthreads();
      float ssW = stepg[ii * 5 + 1 + k];
      if (tid < 64) {
        int r = tid >> 3, j = tid & 7;
        // gWc[r,j] = a_j[r]*s_j/d_j/K ; gWr = (2/(rn2*K)) F^H (E A)
        float d1 = LOG2F_ * (s.tr1[j] + SIGMA2) + 1e-4f;
        float2 gwc = cmul(s.am[j][r], s.sm[j]);
        float invd = 1.0f / (d1 * 4.0f);
        gwc = make_float2(gwc.x * invd, gwc.y * invd);
        float2 acc = make_float2(0, 0);
        for (int i = 0; i < 8; i++)
          acc = cadd(acc, cmul(cconj(s.f[k][8 * r + i]), s.G[8 * r + i][j]));
        float scr = 2.0f / (s.rn2[k] * 4.0f);
        float2 gwr = make_float2(acc.x * scr, acc.y * scr);
        float2 dWc = clampmag(make_float2(ssW * gwc.x, ssW * gwc.y), 0.5f);
        float2 dWr = clampmag(make_float2(ssW * gwr.x, ssW * gwr.y), 0.5f);
        float2 w = s.Wm[k][r][j];
        s.Wm[k][r][j] = make_float2(finz(w.x + dWc.x - dWr.x),
                                    finz(w.y + dWc.y - dWr.y));
      }
      __syncthreads();
      float p  = compute_p(s, k, tid);
      float sc = sqrtf(PT) / (p + 1e-12f);
      if (tid < 64) {
        float2 w = s.Wm[k][tid >> 3][tid & 7];
        s.Wm[k][tid >> 3][tid & 7] = make_float2(finz(w.x * sc), finz(w.y * sc));
      }
      __syncthreads();
    }
    eval_rate_tau(s, Hg, Rg, t, tid, &rates[ii + 1], &taus[ii + 1]);
  }

  // ---- outputs: rates.T (T,3) | taus.T (T,3) | F (K,T,64,8) c64 | W (K,T,8,8) c64 ----
  if (tid == 0) {
#pragma unroll
    for (int it = 0; it < 3; it++) {
      out[t * 3 + it]          = rates[it];
      out[3 * TT + t * 3 + it] = taus[it];
    }
  }
  float* oF = out + 6 * TT;
  for (int i = tid; i < KB * NT * NRF; i += 256) {
    int k = i >> 9, n = (i >> 3) & 63, r = i & 7;
    float2 v = (r == (n >> 3)) ? s.f[k][n] : make_float2(0, 0);
    size_t idx = ((size_t)(k * TT + t) * NT + n) * NRF + r;
    oF[2 * idx]     = v.x;
    oF[2 * idx + 1] = v.y;
  }
  float* oW = out + 6 * TT + 2 * KB * TT * NT * NRF;
  for (int i = tid; i < KB * NRF * MM; i += 256) {
    int k = i >> 6, rm = i & 63;
    float2 v = s.Wm[k][rm >> 3][rm & 7];
    size_t idx = (size_t)(k * TT + t) * (NRF * MM) + rm;
    oW[2 * idx]     = v.x;
    oW[2 * idx + 1] = v.y;
  }
}

}  // namespace

extern "C" void kernel_launch(void* const* d_in, const int* in_sizes, int n_in,
                              void* d_out, int out_size, void* d_ws, size_t ws_size,
                              hipStream_t stream) {
  (void)in_sizes; (void)n_in; (void)d_ws; (void)ws_size; (void)out_size;
  const float2* H  = (const float2*)d_in[0];
  const float2* R  = (const float2*)d_in[1];
  const float2* F0 = (const float2*)d_in[2];
  const float2* W0 = (const float2*)d_in[3];
  const float*  st = (const float*)d_in[4];
  float* out = (float*)d_out;
  hipLaunchKernelGGL(pga_unfold_kernel, dim3(TT), dim3(256), 0, stream,
                     H, R, F0, W0, st, out);
}